// JointGeometryAttention_953482739862
// MI455X (gfx1250) — compile-verified
//
#include <hip/hip_runtime.h>
#include <math.h>

#define NHEADS   16
#define HD       64
#define TSEQ     1024
#define CDIM     1024
#define BATCH    4
#define EPSV     1e-6f
#define KCH      32          // K-chunk staged in LDS per iteration
#define LPAD     36          // LDS row stride (floats): conflict-free + 16B aligned

typedef float v2f __attribute__((ext_vector_type(2)));
typedef float v8f __attribute__((ext_vector_type(8)));

static __device__ __forceinline__ v8f wmma4(v2f a, v2f b, v8f c) {
  // D = A(16x4) * B(4x16) + C, fp32 WMMA (CDNA5)
  return __builtin_amdgcn_wmma_f32_16x16x4_f32(false, a, false, b, (short)0, c,
                                               false, false);
}

static __device__ __forceinline__ float half16_max(float v) {
#pragma unroll
  for (int off = 1; off < 16; off <<= 1) v = fmaxf(v, __shfl_xor(v, off, 16));
  return v;
}
static __device__ __forceinline__ float half16_sum(float v) {
#pragma unroll
  for (int off = 1; off < 16; off <<= 1) v += __shfl_xor(v, off, 16);
  return v;
}
static __device__ __forceinline__ float wave32_sum(float v) {
#pragma unroll
  for (int off = 1; off < 32; off <<= 1) v += __shfl_xor(v, off, 32);
  return v;
}

// ---------------------------------------------------------------------------
// C[M,N] = A[M,K] * B[N,K]^T ; row-major. Block tile 128M x 64N, K-chunk 32,
// double-buffered LDS staging. grid = (N/64, M/128), block = 256 (8 waves).
// Wave w computes rows [128*by + 16w, +16), all 64 N columns (4 accumulators).
// ---------------------------------------------------------------------------
__global__ __launch_bounds__(256) void gemm_abT_wmma(
    const float* __restrict__ A, const float* __restrict__ B,
    float* __restrict__ C, int K, int lda, int ldb, int ldc) {
  __shared__ __align__(16) float lA[2][128 * LPAD];
  __shared__ __align__(16) float lB[2][64 * LPAD];

  const int tid = threadIdx.x;
  const int lane = tid & 31;
  const int wave = tid >> 5;
  const int lm = lane & 15;
  const int lh = lane >> 4;
  const int m0 = blockIdx.y * 128;
  const int n0 = blockIdx.x * 64;

  // cooperative staging map: 256 threads x float4
  const int srow = tid >> 3;         // 0..31
  const int scol = (tid & 7) * 4;    // 0,4,...,28

  const float* Ast = A + (size_t)(m0 + srow) * lda + scol;
  const float* Bst = B + (size_t)(n0 + srow) * ldb + scol;

  float4 ra[4], rb[2];

  // ---- prologue: stage chunk 0 ----
#pragma unroll
  for (int it = 0; it < 4; ++it)
    ra[it] = *(const float4*)(Ast + (size_t)(32 * it) * lda);
#pragma unroll
  for (int it = 0; it < 2; ++it)
    rb[it] = *(const float4*)(Bst + (size_t)(32 * it) * ldb);
#pragma unroll
  for (int it = 0; it < 4; ++it)
    *(float4*)(&lA[0][(srow + 32 * it) * LPAD + scol]) = ra[it];
#pragma unroll
  for (int it = 0; it < 2; ++it)
    *(float4*)(&lB[0][(srow + 32 * it) * LPAD + scol]) = rb[it];
  __syncthreads();

  v8f acc[4] = {v8f{}, v8f{}, v8f{}, v8f{}};
  const int nch = K / KCH;

  for (int c = 0; c < nch; ++c) {
    const int buf = c & 1;

    // prefetch chunk c+1 global -> regs (loads issued before compute)
    if (c + 1 < nch) {
      const int k1 = (c + 1) * KCH;
#pragma unroll
      for (int it = 0; it < 4; ++it)
        ra[it] = *(const float4*)(Ast + (size_t)(32 * it) * lda + k1);
#pragma unroll
      for (int it = 0; it < 2; ++it)
        rb[it] = *(const float4*)(Bst + (size_t)(32 * it) * ldb + k1);
    }
    if (c + 2 < nch) {  // L2 prefetch two chunks ahead (global_prefetch_b8)
      __builtin_prefetch(Ast + (size_t)(c + 2) * KCH, 0, 1);
      __builtin_prefetch(Bst + (size_t)(c + 2) * KCH, 0, 1);
    }

    // ---- compute: preload all fragments from LDS, then 32 WMMAs ----
    const float* pa = &lA[buf][(wave * 16 + lm) * LPAD + 2 * lh];
    const float* pb = &lB[buf][lm * LPAD + 2 * lh];

    v2f af[8];
    v2f bf[8][4];
#pragma unroll
    for (int ks = 0; ks < 8; ++ks) {
      af[ks] = *(const v2f*)(pa + 4 * ks);
#pragma unroll
      for (int j = 0; j < 4; ++j)
        bf[ks][j] = *(const v2f*)(pb + j * 16 * LPAD + 4 * ks);
    }
#pragma unroll
    for (int ks = 0; ks < 8; ++ks)
#pragma unroll
      for (int j = 0; j < 4; ++j)
        acc[j] = wmma4(af[ks], bf[ks][j], acc[j]);

    // stage chunk c+1 regs -> other LDS buffer (disjoint from `buf`)
    if (c + 1 < nch) {
      const int nb = buf ^ 1;
#pragma unroll
      for (int it = 0; it < 4; ++it)
        *(float4*)(&lA[nb][(srow + 32 * it) * LPAD + scol]) = ra[it];
#pragma unroll
      for (int it = 0; it < 2; ++it)
        *(float4*)(&lB[nb][(srow + 32 * it) * LPAD + scol]) = rb[it];
      __syncthreads();
    }
  }

  const int mw = m0 + wave * 16;
#pragma unroll
  for (int j = 0; j < 4; ++j)
#pragma unroll
    for (int r = 0; r < 8; ++r)
      C[(size_t)(mw + r + 8 * lh) * ldc + n0 + j * 16 + lm] = acc[j][r];
}

// ---------------------------------------------------------------------------
// RoPE + norms + layout change. One wave32 per (b,t,h).
// qkv: [B,T,3C]  ->  Q,K,V: [B,H,T,64], q0/k0: [B,H,T]
// ---------------------------------------------------------------------------
__global__ __launch_bounds__(256) void rope_norm_kernel(
    const float* __restrict__ qkv, const float* __restrict__ hyp,
    float* __restrict__ Q, float* __restrict__ Kq, float* __restrict__ V,
    float* __restrict__ q0, float* __restrict__ k0) {
  const int lane = threadIdx.x & 31;
  const int gw = blockIdx.x * 8 + (threadIdx.x >> 5);  // global wave id
  const int h = gw % NHEADS;
  const int t = (gw / NHEADS) % TSEQ;
  const int b = gw / (NHEADS * TSEQ);

  const size_t ibase = ((size_t)(b * TSEQ + t)) * (3 * CDIM) + h * HD;
  const size_t obase = ((size_t)((b * NHEADS + h) * TSEQ) + t) * HD;
  const size_t nidx = (size_t)(b * NHEADS + h) * TSEQ + t;
  const float kc = hyp[h];

  // rotation angle for pair `lane` (pairs are (i, i+32), i = lane)
  const float fr = (float)t * __expf(-((float)(2 * lane) / (float)HD) *
                                     logf(10000.0f));
  float cs, sn;
  __sincosf(fr, &sn, &cs);

  // ---- q ----
  {
    float x1 = qkv[ibase + lane];
    float x2 = qkv[ibase + 32 + lane];
    float o1 = x1 * cs + x2 * sn;
    float o2 = -x1 * sn + x2 * cs;
    Q[obase + lane] = o1;
    Q[obase + 32 + lane] = o2;
    float ss = wave32_sum(o1 * o1 + o2 * o2);
    if (lane == 0) q0[nidx] = sqrtf(kc + ss);
  }
  // ---- k ----
  {
    float x1 = qkv[ibase + CDIM + lane];
    float x2 = qkv[ibase + CDIM + 32 + lane];
    float o1 = x1 * cs + x2 * sn;
    float o2 = -x1 * sn + x2 * cs;
    Kq[obase + lane] = o1;
    Kq[obase + 32 + lane] = o2;
    float ss = wave32_sum(o1 * o1 + o2 * o2);
    if (lane == 0) k0[nidx] = sqrtf(kc + ss);
  }
  // ---- v (copy/transpose) ----
  V[obase + lane] = qkv[ibase + 2 * CDIM + lane];
  V[obase + 32 + lane] = qkv[ibase + 2 * CDIM + 32 + lane];
}

// ---------------------------------------------------------------------------
// Hyperbolic causal attention, flash-style online softmax.
// grid = (B*H, T/128), block = 256 (8 waves); wave handles 16 query rows.
// O written as [B,T,C] (C index = h*64 + d) so the output GEMM reads it flat.
// ---------------------------------------------------------------------------
__global__ __launch_bounds__(256) void hyp_attn_kernel(
    const float* __restrict__ Q, const float* __restrict__ Km,
    const float* __restrict__ V, const float* __restrict__ q0,
    const float* __restrict__ k0, const float* __restrict__ hyp,
    float* __restrict__ O) {
  __shared__ __align__(16) float plds[8][16 * 18];  // per-wave P tile, stride 18

  const int lane = threadIdx.x & 31;
  const int wave = threadIdx.x >> 5;
  const int lm = lane & 15;
  const int lh = lane >> 4;

  const int b = blockIdx.x / NHEADS;
  const int h = blockIdx.x % NHEADS;
  const int l0 = blockIdx.y * 128 + wave * 16;  // first query row of this wave

  const size_t hb = (size_t)(b * NHEADS + h) * TSEQ;
  const float* Qb = Q + hb * HD;
  const float* Kb = Km + hb * HD;
  const float* Vb = V + hb * HD;
  const float* q0b = q0 + hb;
  const float* k0b = k0 + hb;

  const float hk = hyp[h];
  const float inv_hk = 1.0f / hk;
  const float sqk = sqrtf(hk);

  // Preload Q fragments for all 16 K-steps (K dim = 64, step 4).
  v2f qa[16];
#pragma unroll
  for (int d = 0; d < 16; ++d)
    qa[d] = *(const v2f*)(Qb + (size_t)(l0 + lm) * HD + 4 * d + 2 * lh);

  // per-row q0 (rows r+8*lh for r=0..7)
  float q0r[8];
#pragma unroll
  for (int r = 0; r < 8; ++r) q0r[r] = q0b[l0 + r + 8 * lh];

  float m_r[8], l_r[8];
#pragma unroll
  for (int r = 0; r < 8; ++r) { m_r[r] = -1e30f; l_r[r] = 0.0f; }
  v8f o[4] = {v8f{}, v8f{}, v8f{}, v8f{}};

  float* myp = &plds[wave][0];

  for (int s0 = 0; s0 <= l0; s0 += 16) {
    // -------- S = Q * K^T (16x16 over K=64) --------
    v8f s = v8f{};
#pragma unroll
    for (int d = 0; d < 16; ++d) {
      v2f kb = *(const v2f*)(Kb + (size_t)(s0 + lm) * HD + 4 * d + 2 * lh);
      s = wmma4(qa[d], kb, s);
    }

    const float k0c = k0b[s0 + lm];  // per-column k0 (col = lm)

    // -------- hyperbolic weight + causal mask + online softmax --------
#pragma unroll
    for (int r = 0; r < 8; ++r) {
      const int row = l0 + r + 8 * lh;
      const int col = s0 + lm;
      float w;
      if (col > row) {
        w = -INFINITY;
      } else {
        float inner = s[r] - q0r[r] * k0c;
        float arg = fmaxf(-inner * inv_hk, 1.0f + EPSV);
        float dis = sqk * acoshf(arg);
        w = 1.0f / (EPSV + dis);
      }
      float pm = half16_max(w);
      float newm = fmaxf(m_r[r], pm);
      float scale = __expf(m_r[r] - newm);
      float p = __expf(w - newm);  // 0 for masked
      float ps = half16_sum(p);
      l_r[r] = l_r[r] * scale + ps;
      m_r[r] = newm;
      o[0][r] *= scale;
      o[1][r] *= scale;
      o[2][r] *= scale;
      o[3][r] *= scale;
      myp[(r + 8 * lh) * 18 + lm] = p;  // C-layout -> LDS
    }

    // make the per-wave LDS tile visible to our own re-read
    asm volatile("s_wait_dscnt 0" ::: "memory");

    // -------- O += P * V (16x64, K=16 in 4 steps) --------
#pragma unroll
    for (int kk = 0; kk < 4; ++kk) {
      v2f pa = *(const v2f*)(myp + lm * 18 + 4 * kk + 2 * lh);  // A frag of P
      const int krow = s0 + 4 * kk + 2 * lh;
#pragma unroll
      for (int j = 0; j < 4; ++j) {
        v2f vb;
        vb.x = Vb[(size_t)krow * HD + j * 16 + lm];
        vb.y = Vb[(size_t)(krow + 1) * HD + j * 16 + lm];
        o[j] = wmma4(pa, vb, o[j]);
      }
    }
  }

  // -------- normalize and store as [B,T,C] --------
#pragma unroll
  for (int r = 0; r < 8; ++r) {
    const float inv = 1.0f / l_r[r];
    const int t = l0 + r + 8 * lh;
#pragma unroll
    for (int j = 0; j < 4; ++j)
      O[((size_t)(b * TSEQ + t)) * CDIM + h * HD + j * 16 + lm] = o[j][r] * inv;
  }
}

// ---------------------------------------------------------------------------
extern "C" void kernel_launch(void* const* d_in, const int* in_sizes, int n_in,
                              void* d_out, int out_size, void* d_ws,
                              size_t ws_size, hipStream_t stream) {
  (void)in_sizes; (void)n_in; (void)out_size; (void)ws_size;

  const float* x     = (const float*)d_in[0];  // [B,T,C]
  const float* W_qkv = (const float*)d_in[1];  // [3C,C]
  const float* W_out = (const float*)d_in[2];  // [C,C]
  const float* hyp   = (const float*)d_in[3];  // [H]
  float* out = (float*)d_out;                  // [B,T,C]

  const size_t QKV_N = (size_t)BATCH * TSEQ * 3 * CDIM;      // 12.58M floats
  const size_t QN    = (size_t)BATCH * NHEADS * TSEQ * HD;   // 4.19M floats
  const size_t SN    = (size_t)BATCH * NHEADS * TSEQ;        // 65536 floats

  float* ws  = (float*)d_ws;
  float* qkv = ws;
  float* Q   = qkv + QKV_N;
  float* K   = Q + QN;
  float* V   = K + QN;
  float* q0  = V + QN;
  float* k0  = q0 + SN;
  float* AO  = k0 + SN;  // attention output, [B,T,C]

  const int M = BATCH * TSEQ;  // 4096

  // 1) qkv = x @ W_qkv^T   (4096 x 3072 x 1024)
  gemm_abT_wmma<<<dim3((3 * CDIM) / 64, M / 128), 256, 0, stream>>>(
      x, W_qkv, qkv, CDIM, CDIM, CDIM, 3 * CDIM);

  // 2) RoPE + norms + transpose to [B,H,T,64]
  rope_norm_kernel<<<(BATCH * TSEQ * NHEADS) / 8, 256, 0, stream>>>(
      qkv, hyp, Q, K, V, q0, k0);

  // 3) hyperbolic causal attention -> AO [B,T,C]
  hyp_attn_kernel<<<dim3(BATCH * NHEADS, TSEQ / 128), 256, 0, stream>>>(
      Q, K, V, q0, k0, hyp, AO);

  // 4) out = AO @ W_out^T  (4096 x 1024 x 1024)
  gemm_abT_wmma<<<dim3(CDIM / 64, M / 128), 256, 0, stream>>>(
      AO, W_out, out, CDIM, CDIM, CDIM, CDIM);
}